// GCN_4071628996707
// MI455X (gfx1250) — compile-verified
//
#include <hip/hip_runtime.h>
#include <hip/hip_bf16.h>

typedef float v2f __attribute__((ext_vector_type(2)));
typedef float v8f __attribute__((ext_vector_type(8)));

#define D_FEAT 96

// ---------------------------------------------------------------------------
// Zero a float buffer (graph-capture-safe init; harness poisons ws/out).
// ---------------------------------------------------------------------------
__global__ void gcn_zero_f32(float* __restrict__ p, int n) {
    int i = blockIdx.x * blockDim.x + threadIdx.x;
    if (i < n) p[i] = 0.0f;
}

// ---------------------------------------------------------------------------
// In-degree via float atomics: deg[dst[e]] += 1
// ---------------------------------------------------------------------------
__global__ void gcn_degree(const int* __restrict__ dst, float* __restrict__ deg, int E) {
    int e = blockIdx.x * blockDim.x + threadIdx.x;
    if (e < E) {
        int d = __builtin_nontemporal_load(&dst[e]);
        atomicAdd(&deg[d], 1.0f);
    }
}

// ---------------------------------------------------------------------------
// dinv[i] = rsqrt(max(deg[i], 1))  (in place)
// ---------------------------------------------------------------------------
__global__ void gcn_dinv(float* __restrict__ deg, int n) {
    int i = blockIdx.x * blockDim.x + threadIdx.x;
    if (i < n) deg[i] = __frsqrt_rn(fmaxf(deg[i], 1.0f));
}

// ---------------------------------------------------------------------------
// GEMM: Out[N,96] = X[N,96] @ W[96,96]^T + bias, via V_WMMA_F32_16X16X4_F32.
// One wave32 computes a 16x16 output tile; K loop = 24 steps of K=4.
//
// A-matrix (16x4 f32, ISA 7.12.2): lanes 0-15 hold M=0..15, lanes 16-31 hold
//   M=0..15 again; VGPR0/1 hold K = 2*half + {0,1}.
// B-matrix (4x16 f32): mirrored with lane = N index; B[k][n] = W[n][k].
// C/D (16x16 f32): 8 VGPRs; lane<16 -> N=lane, M=r; lane>=16 -> N=lane-16, M=8+r.
// ---------------------------------------------------------------------------
__global__ void gcn_gemm_wmma(const float* __restrict__ X,
                              const float* __restrict__ W,
                              const float* __restrict__ bias,
                              float* __restrict__ Out,
                              int n_rows) {
    const int lane   = threadIdx.x & 31;
    const int wave   = threadIdx.x >> 5;
    const int wpb    = blockDim.x >> 5;
    const int m_tiles = n_rows / 16;           // 3125 (50000 = 3125*16 exactly)
    const int n_tiles = D_FEAT / 16;           // 6
    const int tile   = blockIdx.x * wpb + wave;
    if (tile >= m_tiles * n_tiles) return;

    const int tm   = tile / n_tiles;
    const int tn   = tile % n_tiles;
    const int half = lane >> 4;                // 0: lanes 0-15, 1: lanes 16-31
    const int l16  = lane & 15;

    const int arow = tm * 16 + l16;            // A lane row (M)
    const int bcol = tn * 16 + l16;            // B lane col (N)

    const float* __restrict__ ap = X + arow * D_FEAT + half * 2;
    const float* __restrict__ bp = W + bcol * D_FEAT + half * 2;   // B[k][n] = W[n][k]

    v8f acc = {};
#pragma unroll
    for (int k0 = 0; k0 < D_FEAT; k0 += 4) {
        v2f a; a.x = ap[k0]; a.y = ap[k0 + 1];
        v2f b; b.x = bp[k0]; b.y = bp[k0 + 1];
        acc = __builtin_amdgcn_wmma_f32_16x16x4_f32(
            /*neg_a=*/false, a, /*neg_b=*/false, b,
            /*c_mod=*/(short)0, acc, /*reuse_a=*/false, /*reuse_b=*/false);
    }

    const float bv = bias[bcol];
    const int mbase = tm * 16 + half * 8;
#pragma unroll
    for (int r = 0; r < 8; ++r) {
        Out[(mbase + r) * D_FEAT + bcol] = acc[r] + bv;
    }
}

// ---------------------------------------------------------------------------
// Edge scatter: out[dst] += h[src] * dinv[src]*dinv[dst]
// One edge per 32-lane group; lane l covers features l, l+32, l+64.
// ---------------------------------------------------------------------------
__global__ void gcn_scatter(const int* __restrict__ src,
                            const int* __restrict__ dst,
                            const float* __restrict__ h,
                            const float* __restrict__ dinv,
                            float* __restrict__ out,
                            int E) {
    const int lane = threadIdx.x & 31;
    const int e    = blockIdx.x * (blockDim.x >> 5) + (threadIdx.x >> 5);
    if (e >= E) return;

    const int s = src[e];
    const int d = dst[e];
    const float norm = dinv[s] * dinv[d];

    const float* __restrict__ hs = h + s * D_FEAT;
    float* __restrict__       od = out + d * D_FEAT;
#pragma unroll
    for (int i = 0; i < 3; ++i) {
        const int f = lane + 32 * i;
        atomicAdd(&od[f], hs[f] * norm);   // global_atomic_add_f32, no-return
    }
}

// ---------------------------------------------------------------------------
// Final ReLU over d_out.
// ---------------------------------------------------------------------------
__global__ void gcn_relu(float* __restrict__ p, int n) {
    int i = blockIdx.x * blockDim.x + threadIdx.x;
    if (i < n) p[i] = fmaxf(p[i], 0.0f);
}

// ---------------------------------------------------------------------------
// Launcher.
// Inputs: 0:x[N,96] f32  1:edge_index[2,E] i32  2:W_lin[96,96]  3:b_lin[96]
//         4:W_root[96,96] 5:b_root[96]
// ws layout: h[N*96] f32 | deg/dinv[N] f32   (~19.4 MB)
// ---------------------------------------------------------------------------
extern "C" void kernel_launch(void* const* d_in, const int* in_sizes, int n_in,
                              void* d_out, int out_size, void* d_ws, size_t ws_size,
                              hipStream_t stream) {
    const float* x      = (const float*)d_in[0];
    const int*   eidx   = (const int*)d_in[1];
    const float* W_lin  = (const float*)d_in[2];
    const float* b_lin  = (const float*)d_in[3];
    const float* W_root = (const float*)d_in[4];
    const float* b_root = (const float*)d_in[5];
    float*       out    = (float*)d_out;

    const int N = in_sizes[0] / D_FEAT;     // 50000
    const int E = in_sizes[1] / 2;          // 800000
    const int* src = eidx;                  // edge_index[0]
    const int* dst = eidx + E;              // edge_index[1]

    float* h   = (float*)d_ws;              // N*96
    float* deg = h + (size_t)N * D_FEAT;    // N  (becomes dinv)

    const int T = 256;

    // 1) deg = 0; deg[dst] += 1; dinv = rsqrt(max(deg,1))
    gcn_zero_f32<<<(N + T - 1) / T, T, 0, stream>>>(deg, N);
    gcn_degree<<<(E + T - 1) / T, T, 0, stream>>>(dst, deg, E);
    gcn_dinv<<<(N + T - 1) / T, T, 0, stream>>>(deg, N);

    // 2) h = x @ W_lin^T + b_lin   (WMMA f32)
    const int tiles = (N / 16) * (D_FEAT / 16);
    const int wpb   = T / 32;
    gcn_gemm_wmma<<<(tiles + wpb - 1) / wpb, T, 0, stream>>>(x, W_lin, b_lin, h, N);

    // 3) out = x @ W_root^T + b_root  (initializes d_out)
    gcn_gemm_wmma<<<(tiles + wpb - 1) / wpb, T, 0, stream>>>(x, W_root, b_root, out, N);

    // 4) out[dst] += h[src] * norm   (edge scatter, f32 atomics)
    const int groups_per_block = T / 32;
    gcn_scatter<<<(E + groups_per_block - 1) / groups_per_block, T, 0, stream>>>(
        src, dst, h, deg, out, E);

    // 5) ReLU
    const int total = N * D_FEAT;
    gcn_relu<<<(total + T - 1) / T, T, 0, stream>>>(out, total);
}